// MoE_13477607375000
// MI455X (gfx1250) — compile-verified
//
#include <hip/hip_runtime.h>
#include <hip/hip_bf16.h>

typedef float v2f __attribute__((ext_vector_type(2)));
typedef float v8f __attribute__((ext_vector_type(8)));

#define D_DIM  1024
#define T_TOK  8192
#define TD     (T_TOK * D_DIM)
#define E_EXP  8
#define KC     8          // K chunk staged in LDS
#define WSTR   1040       // padded LDS row stride for W chunk (banks disjoint for half-waves)
#define ASTR   10         // padded LDS row stride for A chunk (keeps b64 8B-aligned)
#define NWAVE  16
#define TPB    512
#define LN_EPS 1e-5f

__device__ __forceinline__ v8f zero8() {
  v8f z = {0.f, 0.f, 0.f, 0.f, 0.f, 0.f, 0.f, 0.f};
  return z;
}

// D = A(16x4) * B(4x16) + C   -- CDNA5 fp32 WMMA
__device__ __forceinline__ v8f wmma4(v2f a, v2f b, v8f c) {
  return __builtin_amdgcn_wmma_f32_16x16x4_f32(false, a, false, b, (short)0, c,
                                               false, false);
}

// ---------------------------------------------------------------------------
// Gating: one wave32 per token. gate = x @ Wg^T + bg; top-2 / bottom-2 softmax
// weights scattered into dense [T, E] arrays (zero for unselected experts).
// ---------------------------------------------------------------------------
__global__ __launch_bounds__(256) void gate_kernel(
    const float* __restrict__ x, const float* __restrict__ Wg,
    const float* __restrict__ bg, float* __restrict__ wtop,
    float* __restrict__ wbot) {
  const int wave = threadIdx.x >> 5;
  const int lane = threadIdx.x & 31;
  const int tok = blockIdx.x * 8 + wave;

  float acc[E_EXP];
#pragma unroll
  for (int e = 0; e < E_EXP; ++e) acc[e] = 0.f;

  const float* xr = x + (size_t)tok * D_DIM;
  for (int d = lane; d < D_DIM; d += 32) {
    float xv = xr[d];
#pragma unroll
    for (int e = 0; e < E_EXP; ++e) acc[e] = fmaf(xv, Wg[e * D_DIM + d], acc[e]);
  }
#pragma unroll
  for (int e = 0; e < E_EXP; ++e) {
#pragma unroll
    for (int m = 16; m >= 1; m >>= 1) acc[e] += __shfl_xor(acc[e], m, 32);
  }

  if (lane == 0) {
    float s[E_EXP];
#pragma unroll
    for (int e = 0; e < E_EXP; ++e) s[e] = acc[e] + bg[e];

    // top-2 (lowest index wins ties, matching top_k order)
    int i1 = 0;
#pragma unroll
    for (int e = 1; e < E_EXP; ++e) if (s[e] > s[i1]) i1 = e;
    int i2 = (i1 == 0) ? 1 : 0;
#pragma unroll
    for (int e = 0; e < E_EXP; ++e) if (e != i1 && s[e] > s[i2]) i2 = e;
    float mt = s[i1];
    float e1 = __expf(s[i1] - mt), e2 = __expf(s[i2] - mt);
    float it = 1.f / (e1 + e2);
    float w1 = e1 * it, w2 = e2 * it;

    // bottom-2
    int j1 = 0;
#pragma unroll
    for (int e = 1; e < E_EXP; ++e) if (s[e] < s[j1]) j1 = e;
    int j2 = (j1 == 0) ? 1 : 0;
#pragma unroll
    for (int e = 0; e < E_EXP; ++e) if (e != j1 && s[e] < s[j2]) j2 = e;
    float mb = fmaxf(s[j1], s[j2]);
    float f1 = __expf(s[j1] - mb), f2 = __expf(s[j2] - mb);
    float ib = 1.f / (f1 + f2);
    float u1 = f1 * ib, u2 = f2 * ib;

#pragma unroll
    for (int e = 0; e < E_EXP; ++e) {
      wtop[(size_t)tok * E_EXP + e] = (e == i1) ? w1 : ((e == i2) ? w2 : 0.f);
      wbot[(size_t)tok * E_EXP + e] = (e == j1) ? u1 : ((e == j2) ? u2 : 0.f);
    }
  }
}

// ---------------------------------------------------------------------------
// One 16xD GEMM tile: A0 is 16 rows (stride D) in global, B0 is DxD weights.
// Each of the 16 waves owns 4 N-tiles of 16 columns. K is chunked through LDS.
// ---------------------------------------------------------------------------
__device__ __forceinline__ void tile_gemm(const float* __restrict__ A0,
                                          const float* __restrict__ B0,
                                          float* sW, float* sA, v8f acc[4],
                                          int tid, int hl, int koff, int nb) {
#pragma unroll
  for (int t = 0; t < 4; ++t) acc[t] = zero8();

  for (int k0 = 0; k0 < D_DIM; k0 += KC) {
    __syncthreads();
    // stage W chunk (KC x D) as float4, padded rows
    {
      const float4* g4 = reinterpret_cast<const float4*>(B0 + (size_t)k0 * D_DIM);
#pragma unroll
      for (int it = 0; it < (KC * D_DIM / 4) / TPB; ++it) {
        int i = tid + it * TPB;           // 0 .. 2047
        int r = i >> 8;                   // chunk row 0..7
        int c4 = i & 255;                 // float4 column
        float4 v = g4[(r << 8) + c4];
        *reinterpret_cast<float4*>(&sW[r * WSTR + (c4 << 2)]) = v;
      }
    }
    // stage A chunk (16 x KC)
    if (tid < 16 * KC) {
      int r = tid >> 3, kk = tid & (KC - 1);
      sA[r * ASTR + kk] = A0[(size_t)r * D_DIM + k0 + kk];
    }
    // prefetch next W chunk into cache (global_prefetch_b8)
    if (k0 + KC < D_DIM)
      __builtin_prefetch(B0 + (size_t)(k0 + KC) * D_DIM + tid * 16, 0, 1);
    __syncthreads();

#pragma unroll
    for (int ks = 0; ks < KC; ks += 4) {
      const float* ap = &sA[hl * ASTR + ks + koff];
      v2f a;
      a.x = ap[0];
      a.y = ap[1];
#pragma unroll
      for (int t = 0; t < 4; ++t) {
        int col = nb + t * 16 + hl;
        v2f b;
        b.x = sW[(ks + koff) * WSTR + col];
        b.y = sW[(ks + koff + 1) * WSTR + col];
        acc[t] = wmma4(a, b, acc[t]);
      }
    }
  }
}

// ---------------------------------------------------------------------------
// Per-token LayerNorm over D across the 16 waves holding the row, applied to
// the WMMA accumulator layout (elem j of lane l => row = j + 8*(l>=16)).
// ---------------------------------------------------------------------------
__device__ __forceinline__ void layernorm_tile(
    v8f acc[4], const float* __restrict__ bias_e, const float* __restrict__ g_e,
    const float* __restrict__ be_e, bool do_relu, float* sSum, float* sSq,
    float* sMean, float* sRstd, int tid, int wave, int hl, int hi, int nb) {
#pragma unroll
  for (int t = 0; t < 4; ++t) {
    float bv = bias_e[nb + t * 16 + hl];
#pragma unroll
    for (int j = 0; j < 8; ++j) acc[t][j] += bv;
  }
  float s[8], q[8];
#pragma unroll
  for (int j = 0; j < 8; ++j) {
    float a0 = acc[0][j], a1 = acc[1][j], a2 = acc[2][j], a3 = acc[3][j];
    s[j] = (a0 + a1) + (a2 + a3);
    q[j] = fmaf(a0, a0, fmaf(a1, a1, fmaf(a2, a2, a3 * a3)));
  }
#pragma unroll
  for (int m = 8; m >= 1; m >>= 1) {
#pragma unroll
    for (int j = 0; j < 8; ++j) {
      s[j] += __shfl_xor(s[j], m, 32);
      q[j] += __shfl_xor(q[j], m, 32);
    }
  }
  if (hl == 0) {
#pragma unroll
    for (int j = 0; j < 8; ++j) {
      sSum[wave * 16 + hi * 8 + j] = s[j];
      sSq[wave * 16 + hi * 8 + j] = q[j];
    }
  }
  __syncthreads();
  if (tid < 16) {
    float tot = 0.f, tq = 0.f;
#pragma unroll
    for (int w = 0; w < NWAVE; ++w) {
      tot += sSum[w * 16 + tid];
      tq += sSq[w * 16 + tid];
    }
    float mean = tot * (1.f / D_DIM);
    float var = fmaxf(tq * (1.f / D_DIM) - mean * mean, 0.f);
    sMean[tid] = mean;
    sRstd[tid] = rsqrtf(var + LN_EPS);
  }
  __syncthreads();
#pragma unroll
  for (int t = 0; t < 4; ++t) {
    int col = nb + t * 16 + hl;
    float gv = g_e[col], bev = be_e[col];
#pragma unroll
    for (int j = 0; j < 8; ++j) {
      int row = hi * 8 + j;
      float v = fmaf((acc[t][j] - sMean[row]) * sRstd[row], gv, bev);
      if (do_relu) v = fmaxf(v, 0.f);
      acc[t][j] = v;
    }
  }
}

// ---------------------------------------------------------------------------
// Fused dense-MoE FFN: per 16-token tile, loop experts:
//   fc1 (WMMA) -> LN -> relu -> stage h -> fc2 (WMMA) -> LN -> top/bot combine
// h stages through the (not-yet-final) output_bottom slice of d_out (L2-hot).
// ---------------------------------------------------------------------------
__global__ __launch_bounds__(TPB) void moe_ffn_kernel(
    const float* __restrict__ x, const float* __restrict__ W1,
    const float* __restrict__ b1, const float* __restrict__ g1,
    const float* __restrict__ be1, const float* __restrict__ W2,
    const float* __restrict__ b2, const float* __restrict__ g2,
    const float* __restrict__ be2, const float* __restrict__ wtop,
    const float* __restrict__ wbot, float* __restrict__ out,
    float* __restrict__ lossb) {
  __shared__ __align__(16) float sW[KC * WSTR];
  __shared__ __align__(16) float sA[16 * ASTR];
  __shared__ float sSum[NWAVE * 16];
  __shared__ float sSq[NWAVE * 16];
  __shared__ float sMean[16], sRstd[16];
  __shared__ float sWt[16], sWb[16];

  const int tid = threadIdx.x;
  const int wave = tid >> 5;
  const int lane = tid & 31;
  const int hl = lane & 15;   // N position inside 16-wide tile
  const int hi = lane >> 4;   // half-wave selects K-pair / row-half
  const int koff = hi * 2;
  const int nb = wave * 64;   // 4 N-tiles per wave
  const int tb = blockIdx.x * 16;

  float* stage = out + (size_t)2 * TD + (size_t)tb * D_DIM;  // h staging
  const float* xA = x + (size_t)tb * D_DIM;

  v8f otop[4], obot[4], acc[4];
#pragma unroll
  for (int t = 0; t < 4; ++t) {
    otop[t] = zero8();
    obot[t] = zero8();
  }

  for (int e = 0; e < E_EXP; ++e) {
    __syncthreads();
    if (tid < 16) {
      sWt[tid] = wtop[(size_t)(tb + tid) * E_EXP + e];
      sWb[tid] = wbot[(size_t)(tb + tid) * E_EXP + e];
    }

    // ---- fc1: h = x @ W1[e] + b1[e], LN, relu ----
    tile_gemm(xA, W1 + (size_t)e * D_DIM * D_DIM, sW, sA, acc, tid, hl, koff, nb);
    layernorm_tile(acc, b1 + e * D_DIM, g1 + e * D_DIM, be1 + e * D_DIM, true,
                   sSum, sSq, sMean, sRstd, tid, wave, hl, hi, nb);

    // stage h_norm for fc2 (this block's private 16xD slice)
#pragma unroll
    for (int t = 0; t < 4; ++t) {
      int col = nb + t * 16 + hl;
#pragma unroll
      for (int j = 0; j < 8; ++j) {
        int row = hi * 8 + j;
        stage[(size_t)row * D_DIM + col] = acc[t][j];
      }
    }
    __threadfence();
    __syncthreads();

    // ---- fc2: o = h @ W2[e] + b2[e], LN ----
    tile_gemm(stage, W2 + (size_t)e * D_DIM * D_DIM, sW, sA, acc, tid, hl, koff, nb);
    layernorm_tile(acc, b2 + e * D_DIM, g2 + e * D_DIM, be2 + e * D_DIM, false,
                   sSum, sSq, sMean, sRstd, tid, wave, hl, hi, nb);

    // ---- gated accumulate ----
#pragma unroll
    for (int t = 0; t < 4; ++t) {
#pragma unroll
      for (int j = 0; j < 8; ++j) {
        int row = hi * 8 + j;
        otop[t][j] = fmaf(sWt[row], acc[t][j], otop[t][j]);
        obot[t][j] = fmaf(sWb[row], acc[t][j], obot[t][j]);
      }
    }
  }

  // ---- per-batch ||output_top - output_bottom||^2 partial ----
  float ls = 0.f;
#pragma unroll
  for (int t = 0; t < 4; ++t) {
#pragma unroll
    for (int j = 0; j < 8; ++j) {
      float d = otop[t][j] - obot[t][j];
      ls = fmaf(d, d, ls);
    }
  }
#pragma unroll
  for (int m = 16; m >= 1; m >>= 1) ls += __shfl_xor(ls, m, 32);
  if (lane == 0) sSum[wave] = ls;
  __syncthreads();
  if (tid == 0) {
    float tot = 0.f;
#pragma unroll
    for (int w = 0; w < NWAVE; ++w) tot += sSum[w];
    atomicAdd(&lossb[tb >> 10], tot);
  }

  // ---- final outputs: output = otop + x, output_top, output_bottom ----
#pragma unroll
  for (int t = 0; t < 4; ++t) {
    int col = nb + t * 16 + hl;
#pragma unroll
    for (int j = 0; j < 8; ++j) {
      int row = hi * 8 + j;
      size_t idx = (size_t)(tb + row) * D_DIM + col;
      float ot = otop[t][j], ob = obot[t][j];
      out[idx] = ot + x[idx];
      out[(size_t)TD + idx] = ot;
      out[(size_t)2 * TD + idx] = ob;
    }
  }
}

__global__ void zero_loss_kernel(float* __restrict__ lossb) {
  if (threadIdx.x < 8) lossb[threadIdx.x] = 0.f;
}

__global__ void loss_final_kernel(const float* __restrict__ lossb,
                                  float* __restrict__ out) {
  int t = threadIdx.x;
  float v = 0.f;
  if (t < 8) v = 1.f / (sqrtf(lossb[t]) + 1e-8f);
#pragma unroll
  for (int m = 16; m >= 1; m >>= 1) v += __shfl_xor(v, m, 32);
  if (t == 0) out[(size_t)3 * TD] = v * (1.f / 8.f);
}

extern "C" void kernel_launch(void* const* d_in, const int* in_sizes, int n_in,
                              void* d_out, int out_size, void* d_ws,
                              size_t ws_size, hipStream_t stream) {
  const float* x   = (const float*)d_in[0];
  const float* Wg  = (const float*)d_in[1];
  const float* bg  = (const float*)d_in[2];
  const float* W1  = (const float*)d_in[3];
  const float* b1  = (const float*)d_in[4];
  const float* g1  = (const float*)d_in[5];
  const float* be1 = (const float*)d_in[6];
  const float* W2  = (const float*)d_in[7];
  const float* b2  = (const float*)d_in[8];
  const float* g2  = (const float*)d_in[9];
  const float* be2 = (const float*)d_in[10];
  float* out = (float*)d_out;

  float* wtop  = (float*)d_ws;                       // T*E
  float* wbot  = wtop + (size_t)T_TOK * E_EXP;       // T*E
  float* lossb = wbot + (size_t)T_TOK * E_EXP;       // 8 per-batch partials

  zero_loss_kernel<<<1, 32, 0, stream>>>(lossb);
  gate_kernel<<<T_TOK / 8, 256, 0, stream>>>(x, Wg, bg, wtop, wbot);
  moe_ffn_kernel<<<T_TOK / 16, TPB, 0, stream>>>(x, W1, b1, g1, be1, W2, b2, g2,
                                                 be2, wtop, wbot, out, lossb);
  loss_final_kernel<<<1, 32, 0, stream>>>(lossb, out);
}